// _ConvBlock2_43018392436822
// MI455X (gfx1250) — compile-verified
//
#include <hip/hip_runtime.h>

typedef __attribute__((ext_vector_type(2))) float v2f;
typedef __attribute__((ext_vector_type(8))) float v8f;

// ---------------- zero init ----------------
__global__ void zero_kernel(float* __restrict__ p, int n) {
  int i = blockIdx.x * blockDim.x + threadIdx.x;
  if (i < n) p[i] = 0.0f;
}

// ---------------- weighted scatter-add ----------------
// out[dst[e]*64 + c] += attr[e] * in[src[e]*in_stride + c_off + c]
// 16 lanes per edge, float4 per lane -> 4 fp32 global atomics (no-return).
__global__ void scatter_kernel(const float* __restrict__ in,
                               const float* __restrict__ attr,
                               const int* __restrict__ src,
                               const int* __restrict__ dst,
                               float* __restrict__ out,
                               int nedges, int in_stride, int c_off) {
  long t = (long)blockIdx.x * blockDim.x + threadIdx.x;
  int e = (int)(t >> 4);
  if (e >= nedges) return;
  int c = ((int)t & 15) * 4;
  float a = attr[e];
  const float* ip = in + (long)src[e] * in_stride + c_off + c;
  float4 v = *(const float4*)ip;
  float* op = out + (long)dst[e] * 64 + c;
  atomicAdd(op + 0, a * v.x);
  atomicAdd(op + 1, a * v.y);
  atomicAdd(op + 2, a * v.z);
  atomicAdd(op + 3, a * v.w);
}

// ---------------- out = t[:, 0:64] + bias ----------------
__global__ void init_kernel(const float* __restrict__ t,
                            const float* __restrict__ bias,
                            float* __restrict__ out, int nrows) {
  long i = (long)blockIdx.x * blockDim.x + threadIdx.x;
  int row = (int)(i >> 4);
  if (row >= nrows) return;
  int c = ((int)i & 15) * 4;
  float4 v = *(const float4*)(t + (long)row * 128 + c);
  float4 b = *(const float4*)(bias + c);
  float4 r;
  r.x = v.x + b.x; r.y = v.y + b.y; r.z = v.z + b.z; r.w = v.w + b.w;
  *(float4*)(out + (long)row * 64 + c) = r;
}

// ---------------- fused dense GEMM: T = A @ [Wroot | Wnbr] ----------------
// A: [M][64] fp32, T: [M][128] fp32.  V_WMMA_F32_16X16X4_F32 path.
// 256 threads = 8 waves; wave w owns M-tile (blockIdx.x*8 + w): 16 rows x 128 cols.
// LDS holds W packed in K-pairs: Wp[p][col][j] = W[2p+j][col]  (p=0..31, col=0..127)
// so every WMMA B fragment is one contiguous ds_load_b64 into an aligned pair.
__global__ void __launch_bounds__(256)
gemm_kernel(const float* __restrict__ A,
            const float* __restrict__ Wroot,
            const float* __restrict__ Wnbr,
            float* __restrict__ T, int M) {
  __shared__ float Wld[32 * 128 * 2];  // 32 KB
  int tid = threadIdx.x;

  // cooperative fill with K-pair packing: W[k][c] -> Wld[(k>>1)*256 + c*2 + (k&1)]
  for (int i = tid * 4; i < 64 * 64; i += 256 * 4) {
    int k = i >> 6;
    int c = i & 63;
    int base = (k >> 1) * 256 + (k & 1);
    float4 vr = *(const float4*)&Wroot[i];
    float4 vn = *(const float4*)&Wnbr[i];
    Wld[base + (c + 0) * 2] = vr.x;
    Wld[base + (c + 1) * 2] = vr.y;
    Wld[base + (c + 2) * 2] = vr.z;
    Wld[base + (c + 3) * 2] = vr.w;
    Wld[base + (c + 64 + 0) * 2] = vn.x;
    Wld[base + (c + 64 + 1) * 2] = vn.y;
    Wld[base + (c + 64 + 2) * 2] = vn.z;
    Wld[base + (c + 64 + 3) * 2] = vn.w;
  }
  __syncthreads();

  int wave = tid >> 5;
  int lane = tid & 31;
  int mtile = blockIdx.x * 8 + wave;
  if (mtile * 16 >= M) return;  // wave-uniform: EXEC stays all-1s for WMMA waves

  int row   = mtile * 16 + (lane & 15);
  int khalf = (lane >> 4) * 2;  // lanes 0-15 -> K k,k+1 ; lanes 16-31 -> K k+2,k+3
  int ncol  = lane & 15;

  v8f acc[8] = {};
  const float* arow = A + (long)row * 64;

  #pragma unroll 4
  for (int k = 0; k < 64; k += 4) {
    // A fragment: contiguous pair -> one global_load_b64
    v2f a = *(const v2f*)&arow[k + khalf];
    int p = (k + khalf) >> 1;  // packed K-pair index
    const float* bbase = &Wld[p * 256 + ncol * 2];
    #pragma unroll
    for (int n = 0; n < 8; n++) {
      // B fragment: contiguous pair -> one ds_load_b64 into an aligned VGPR pair
      v2f b = *(const v2f*)(bbase + n * 32);
      acc[n] = __builtin_amdgcn_wmma_f32_16x16x4_f32(
          false, a, false, b, (short)0, acc[n], false, false);
    }
  }

  // D layout: lane<16 -> rows mtile*16 + r (r=0..7); lane>=16 -> rows +8
  int rbase = mtile * 16 + (lane >> 4) * 8;
  #pragma unroll
  for (int n = 0; n < 8; n++) {
    #pragma unroll
    for (int r = 0; r < 8; r++) {
      T[(long)(rbase + r) * 128 + n * 16 + ncol] = acc[n][r];
    }
  }
}

extern "C" void kernel_launch(void* const* d_in, const int* in_sizes, int n_in,
                              void* d_out, int out_size, void* d_ws, size_t ws_size,
                              hipStream_t stream) {
  const float* x         = (const float*)d_in[0];
  const float* pool_attr = (const float*)d_in[1];
  const float* pp_attr   = (const float*)d_in[2];
  const float* W1_root   = (const float*)d_in[3];
  const float* W1_nbr    = (const float*)d_in[4];
  const float* b1        = (const float*)d_in[5];
  const float* W2_root   = (const float*)d_in[6];
  const float* W2_nbr    = (const float*)d_in[7];
  const float* b2        = (const float*)d_in[8];
  const int*   pool_src  = (const int*)d_in[9];
  const int*   pool_dst  = (const int*)d_in[10];
  const int*   pp_src    = (const int*)d_in[11];
  const int*   pp_dst    = (const int*)d_in[12];

  const int E_pool = in_sizes[1];
  const int E_pp   = in_sizes[2];
  const int M      = out_size / 64;  // n_coarse

  float* ws     = (float*)d_ws;
  float* pooled = ws;                          // M*64
  float* t      = pooled + (size_t)M * 64;     // M*128
  float* h      = t + (size_t)M * 128;         // M*64
  float* out    = (float*)d_out;

  const int nelem = M * 64;
  const int gemm_grid = (M + 127) / 128;
  const long pool_threads = (long)E_pool * 16;
  const long pp_threads   = (long)E_pp * 16;
  const long init_threads = (long)M * 16;

  // 1) pooled = 0
  zero_kernel<<<(nelem + 255) / 256, 256, 0, stream>>>(pooled, nelem);
  // 2) pooled += pool_attr * x[pool_src]  (scatter to pool_dst)
  scatter_kernel<<<(int)((pool_threads + 255) / 256), 256, 0, stream>>>(
      x, pool_attr, pool_src, pool_dst, pooled, E_pool, 64, 0);
  // 3) t = pooled @ [W1_root | W1_nbr]
  gemm_kernel<<<gemm_grid, 256, 0, stream>>>(pooled, W1_root, W1_nbr, t, M);
  // 4) h = t_root + b1
  init_kernel<<<(int)((init_threads + 255) / 256), 256, 0, stream>>>(t, b1, h, M);
  // 5) h[dst] += pp_attr * t_nbr[src]
  scatter_kernel<<<(int)((pp_threads + 255) / 256), 256, 0, stream>>>(
      t, pp_attr, pp_src, pp_dst, h, E_pp, 128, 64);
  // 6) t = h @ [W2_root | W2_nbr]
  gemm_kernel<<<gemm_grid, 256, 0, stream>>>(h, W2_root, W2_nbr, t, M);
  // 7) out = t_root + b2
  init_kernel<<<(int)((init_threads + 255) / 256), 256, 0, stream>>>(t, b2, out, M);
  // 8) out[dst] += pp_attr * t_nbr[src]
  scatter_kernel<<<(int)((pp_threads + 255) / 256), 256, 0, stream>>>(
      t, pp_attr, pp_src, pp_dst, out, E_pp, 128, 64);
}